// LIF_52707838656664
// MI455X (gfx1250) — compile-verified
//
#include <hip/hip_runtime.h>
#include <stdint.h>

// LIF forward, x/y: [T=16, B*C*H*W = 4,194,304] f32.
// Pure HBM-streaming kernel (~536 MB total traffic -> ~23 us at 23.3 TB/s).
// CDNA5 path: per-lane GLOBAL_LOAD_ASYNC_TO_LDS_B128 staging (ASYNCcnt),
// staged s_wait_asynccnt, ds_load_b128 consumption, NT b128 stores.

typedef float v4f __attribute__((ext_vector_type(4)));

#define T_STEPS 16
#define LIF_V_TH 1.0f
#define LIF_TAU  0.5f
#define BLOCK    256

__global__ __launch_bounds__(BLOCK) void lif_fwd_kernel(
    const float* __restrict__ x,   // [T][per_t] f32
    float* __restrict__ y,         // [T][per_t] f32
    unsigned per_t_vec)            // per-timestep element count / 4
{
    // 16 timesteps x 256 threads x 16B = 64 KB staging buffer
    __shared__ v4f smem[T_STEPS * BLOCK];

    const unsigned tid = threadIdx.x;
    const unsigned nv  = blockIdx.x * BLOCK + tid;   // float4 index within a timestep
    if (nv >= per_t_vec) return;

    // Per-lane byte offset within one timestep slab (< 2^27, fits signed VADDR_I32).
    const unsigned byteOff = nv * 16u;

    // LDS byte offset of this thread's t=0 slot: low 32 bits of the generic
    // pointer to a __shared__ object are the LDS offset (flat-aperture rule).
    const unsigned lds0 = (unsigned)(uintptr_t)&smem[tid];

    // ---- Issue all 16 async global->LDS B128 copies up front (deep MLP). ----
    // GVS addressing: mem = SGPR64(base + t*slab) + VGPR_I32(byteOff).
    // The 16MB timestep stride lives in the uniform SGPR base, so the per-lane
    // VGPR offset never overflows.
#pragma unroll
    for (int t = 0; t < T_STEPS; ++t) {
        const float* src = x + (size_t)t * (size_t)per_t_vec * 4u;
        const unsigned ldsOff = lds0 + (unsigned)t * (BLOCK * 16u);
        asm volatile("global_load_async_to_lds_b128 %0, %1, %2"
                     :: "v"(ldsOff), "v"(byteOff), "s"(src)
                     : "memory");
    }

    v4f v;
    v[0] = 0.0f; v[1] = 0.0f; v[2] = 0.0f; v[3] = 0.0f;
    v4f* __restrict__ out = (v4f*)y;

    // One LIF step: m = v*tau + x ; y = (m > v_th) ; v = spike ? 0 : m.
    // (m - 1.0f > 0) == (m > 1.0f) exactly in f32 (Sterbenz around 1.0).
#define LIF_STEP(t)                                                           \
    {                                                                         \
        v4f xv = smem[(t) * BLOCK + tid]; /* ds_load_b128 */                  \
        v4f yv;                                                               \
        _Pragma("unroll")                                                     \
        for (int i = 0; i < 4; ++i) {                                         \
            float m = __builtin_fmaf(v[i], LIF_TAU, xv[i]);                   \
            bool  s = m > LIF_V_TH;                                           \
            yv[i]   = s ? 1.0f : 0.0f;                                        \
            v[i]    = s ? 0.0f : m;                                           \
        }                                                                     \
        __builtin_nontemporal_store(                                          \
            yv, out + (size_t)(t) * (size_t)per_t_vec + nv);                  \
    }

    // Async loads complete in order -> count-staged waits overlap the scan
    // with the remaining in-flight copies. Only the issuing lane reads its
    // own LDS slots, so no workgroup barrier is required.
    asm volatile("s_wait_asynccnt 12" ::: "memory");
    LIF_STEP(0)  LIF_STEP(1)  LIF_STEP(2)  LIF_STEP(3)
    asm volatile("s_wait_asynccnt 8" ::: "memory");
    LIF_STEP(4)  LIF_STEP(5)  LIF_STEP(6)  LIF_STEP(7)
    asm volatile("s_wait_asynccnt 4" ::: "memory");
    LIF_STEP(8)  LIF_STEP(9)  LIF_STEP(10) LIF_STEP(11)
    asm volatile("s_wait_asynccnt 0" ::: "memory");
    LIF_STEP(12) LIF_STEP(13) LIF_STEP(14) LIF_STEP(15)

#undef LIF_STEP
}

extern "C" void kernel_launch(void* const* d_in, const int* in_sizes, int n_in,
                              void* d_out, int out_size, void* d_ws, size_t ws_size,
                              hipStream_t stream) {
    (void)n_in; (void)out_size; (void)d_ws; (void)ws_size;

    const float* x = (const float*)d_in[0];
    float*       y = (float*)d_out;

    // x is [T=16, B, C, H, W]; per-timestep element count:
    const unsigned per_t     = (unsigned)(in_sizes[0] / T_STEPS);  // 4,194,304
    const unsigned per_t_vec = per_t / 4u;                          // 1,048,576? no: 1,048,576*4=4.19M -> 1,048,576
    const unsigned blocks    = (per_t_vec + BLOCK - 1) / BLOCK;

    lif_fwd_kernel<<<blocks, BLOCK, 0, stream>>>(x, y, per_t_vec);
}